// MeanShiftDropout_54580444397811
// MI455X (gfx1250) — compile-verified
//
#include <hip/hip_runtime.h>

typedef __attribute__((ext_vector_type(2))) float v2f;
typedef __attribute__((ext_vector_type(4))) float v4f;
typedef __attribute__((ext_vector_type(8))) float v8f;

// Problem dims (fixed by reference setup_inputs)
#define B_DIM 256
#define C_DIM 1024
#define D_DIM 128
#define NSUB 8
#define SUBB 32
#define GROUPS 8
#define CPG 128
#define OUTCH 2048
#define OUTPER 256
#define NELEM ((size_t)B_DIM * C_DIM * D_DIM)   // 33554432
#define N4 (NELEM / 4)                          // 8388608 float4s

// ---------------- Phase 0: zero the atomic accumulators ----------------
__global__ void zero_kernel(float* __restrict__ sum_ws, float* __restrict__ cnt_ws) {
    int t = blockIdx.x * blockDim.x + threadIdx.x;
    if (t < OUTCH) { sum_ws[t] = 0.0f; cnt_ws[t] = 0.0f; }
}

// ---------------- Phase 1: WMMA reduction over (sub-batch, D) ----------------
// grid = (C_DIM/16, NSUB), block = 32 (one wave).
// Wave w computes value[s, c0..c0+15] = sum over 32 batch rows x 128 D of x.
// A-tile (16x4 f32): lane<16 holds A[m][0..1], lane>=16 holds A[m][2..3], m=lane&15.
// B = all-ones 4x16  =>  D[m][n] = row-sum of A (replicated over n), accumulated in C.
__global__ void reduce_kernel(const float* __restrict__ x, float* __restrict__ value_ws) {
    const int lane  = threadIdx.x;
    const int m     = lane & 15;
    const int khalf = lane >> 4;          // 0: K={0,1}  1: K={2,3}
    const int c0    = blockIdx.x * 16;
    const int s     = blockIdx.y;

    // base points at x[s*32, c0+m, 2*khalf]
    const float* base = x + ((size_t)(s * SUBB) * C_DIM + (c0 + m)) * D_DIM + 2 * khalf;

    v2f bones; bones[0] = 1.0f; bones[1] = 1.0f;
    v8f acc0 = {};
    v8f acc1 = {};

    for (int b = 0; b < SUBB; ++b) {
        const float* rowp = base + (size_t)b * (C_DIM * D_DIM);
#pragma unroll
        for (int d0 = 0; d0 < D_DIM; d0 += 8) {
            v2f a0 = *(const v2f*)(rowp + d0);       // covers K = d0 .. d0+3 across lanes
            v2f a1 = *(const v2f*)(rowp + d0 + 4);   // covers K = d0+4 .. d0+7
            acc0 = __builtin_amdgcn_wmma_f32_16x16x4_f32(
                false, a0, false, bones, (short)0, acc0, false, false);
            acc1 = __builtin_amdgcn_wmma_f32_16x16x4_f32(
                false, a1, false, bones, (short)0, acc1, false, false);
        }
    }

    v8f acc = acc0 + acc1;
    // C/D layout: VGPR j, lanes 0-15 -> M=j; lanes 16-31 -> M=8+j (all columns equal).
    if (lane == 0) {
#pragma unroll
        for (int j = 0; j < 8; ++j) value_ws[(size_t)s * C_DIM + c0 + j] = acc[j];
    }
    if (lane == 16) {
#pragma unroll
        for (int j = 0; j < 8; ++j) value_ws[(size_t)s * C_DIM + c0 + 8 + j] = acc[j];
    }
}

// ---------------- Phase 2a: segment-sum scatter (8192 entries) ----------------
__global__ void scatter_kernel(const float* __restrict__ value_ws,
                               const long long* __restrict__ index,
                               float* __restrict__ sum_ws, float* __restrict__ cnt_ws) {
    int t = blockIdx.x * blockDim.x + threadIdx.x;   // t = s*1024 + c
    if (t >= NSUB * C_DIM) return;
    int s  = t >> 10;
    int c  = t & 1023;
    int g  = c >> 7;
    int cc = c & 127;
    int o  = g * OUTPER + (int)index[((size_t)s * GROUPS + g) * CPG + cc];
    atomicAdd(&sum_ws[o], value_ws[t]);
    atomicAdd(&cnt_ws[o], (float)SUBB);
}

// ---------------- Phase 2b: mean = sum/((cnt+eps)*D) - bias ----------------
__global__ void mean_kernel(const float* __restrict__ sum_ws, const float* __restrict__ cnt_ws,
                            const float* __restrict__ bias, float* __restrict__ mean_ws) {
    int t = blockIdx.x * blockDim.x + threadIdx.x;
    if (t < OUTCH)
        mean_ws[t] = sum_ws[t] / ((cnt_ws[t] + 1e-10f) * (float)D_DIM) - bias[t];
}

// ---------------- Phase 2c: gather per-(s,c) shift values ----------------
__global__ void mval_kernel(const float* __restrict__ mean_ws,
                            const long long* __restrict__ index,
                            float* __restrict__ mval_ws) {
    int t = blockIdx.x * blockDim.x + threadIdx.x;   // t = s*1024 + c
    if (t >= NSUB * C_DIM) return;
    int s  = t >> 10;
    int c  = t & 1023;
    int g  = c >> 7;
    int cc = c & 127;
    mval_ws[t] = mean_ws[g * OUTPER + (int)index[((size_t)s * GROUPS + g) * CPG + cc]];
}

// ---------------- Phase 3: fused streaming subtract ----------------
// One wave per 128-float row; each lane handles one float4 (b128 traffic).
// x comes back out of the 192MB L2 (loaded RT in phase 1); lower/upper and all
// stores are non-temporal so 640 MiB of single-use traffic doesn't evict x.
__global__ void apply_kernel(const float* __restrict__ x, const float* __restrict__ lo,
                             const float* __restrict__ up, const float* __restrict__ mval_ws,
                             float* __restrict__ out) {
    size_t t = (size_t)blockIdx.x * blockDim.x + threadIdx.x;  // float4 index
    size_t row = t >> 5;                 // 32 float4s per (b,c) row
    int c = (int)(row & 1023);
    int b = (int)(row >> 10);
    float m = mval_ws[((size_t)(b >> 5) << 10) + c];   // wave-uniform -> broadcast load

    const v4f* x4 = (const v4f*)x;
    const v4f* l4 = (const v4f*)lo;
    const v4f* u4 = (const v4f*)up;
    v4f* o4 = (v4f*)out;

    v4f xv = x4[t];                                   // RT: expect L2 hit
    v4f lv = __builtin_nontemporal_load(l4 + t);      // single-use
    v4f uv = __builtin_nontemporal_load(u4 + t);      // single-use
    v4f mm = { m, m, m, m };

    __builtin_nontemporal_store(xv - mm, o4 + t);
    __builtin_nontemporal_store(lv - mm, o4 + N4 + t);
    __builtin_nontemporal_store(uv - mm, o4 + 2 * N4 + t);
}

extern "C" void kernel_launch(void* const* d_in, const int* in_sizes, int n_in,
                              void* d_out, int out_size, void* d_ws, size_t ws_size,
                              hipStream_t stream) {
    const float*     x     = (const float*)d_in[0];
    const float*     lower = (const float*)d_in[1];
    const float*     upper = (const float*)d_in[2];
    const long long* index = (const long long*)d_in[3];
    const float*     bias  = (const float*)d_in[4];
    float*           out   = (float*)d_out;

    // Workspace layout (floats): value[8192] | sum[2048] | cnt[2048] | mean[2048] | mval[8192]
    float* ws       = (float*)d_ws;
    float* value_ws = ws;
    float* sum_ws   = ws + 8192;
    float* cnt_ws   = ws + 10240;
    float* mean_ws  = ws + 12288;
    float* mval_ws  = ws + 14336;

    zero_kernel   <<<(OUTCH + 255) / 256, 256, 0, stream>>>(sum_ws, cnt_ws);
    reduce_kernel <<<dim3(C_DIM / 16, NSUB), 32, 0, stream>>>(x, value_ws);
    scatter_kernel<<<(NSUB * C_DIM + 255) / 256, 256, 0, stream>>>(value_ws, index, sum_ws, cnt_ws);
    mean_kernel   <<<(OUTCH + 255) / 256, 256, 0, stream>>>(sum_ws, cnt_ws, bias, mean_ws);
    mval_kernel   <<<(NSUB * C_DIM + 255) / 256, 256, 0, stream>>>(mean_ws, index, mval_ws);
    apply_kernel  <<<(unsigned)(N4 / 256), 256, 0, stream>>>(x, lower, upper, mval_ws, out);
}